// GNNStack_71262097375399
// MI455X (gfx1250) — compile-verified
//
#include <hip/hip_runtime.h>
#include <hip/hip_bf16.h>
#include <stdint.h>

// ---------------------------------------------------------------------------
// CDNA5 WMMA (wave32).  f32 16x16x4: A=16x4 (2 VGPR), B=4x16 (2 VGPR),
// C/D = 16x16 f32 (8 VGPR).
// ---------------------------------------------------------------------------
typedef __attribute__((ext_vector_type(2))) float v2f;
typedef __attribute__((ext_vector_type(8))) float v8f;

#define LN_EPS 1e-5f

// ---------------------------------------------------------------------------
// fp32 GEMM via V_WMMA_F32_16X16X4_F32.  Fully static shapes: K and NCOLS are
// template parameters so every access is base + compile-time immediate offset.
// One wave computes an (MB*16) x 16 strip of C = op(A) @ B [+ bias], with an
// explicit 2-deep load pipeline (loads of step kk+1 issued before WMMAs of kk).
// REQUIRES: M % (MB*16) == 0, NCOLS % 16 == 0, K % 4 == 0.
//   A: M x K row-major   (ReLU'd on load when RELU)
//   B: K x NCOLS row-major
//   C: M x NCOLS row-major
// Layouts per CDNA5 ISA:
//   A frag: lanes 0-15 row M=lane&15, VGPR j holds K = k + 2*(lane>=16) + j
//   B frag: VGPR j holds row K = k + 2*(lane>=16) + j, striped over cols
//   C/D:    VGPR r -> (m0 + r + 8*(lane>=16), n0 + (lane&15))
// ---------------------------------------------------------------------------
template <int MB, int K, int NCOLS, bool RELU>
__global__ void wmma_gemm_f32(const float* __restrict__ A,
                              const float* __restrict__ B,
                              const float* __restrict__ bias,
                              float* __restrict__ C) {
  const int lane = threadIdx.x & 31;
  const int half = lane >> 4;
  const int l16  = lane & 15;
  const int m0 = blockIdx.x * (MB * 16);
  const int n0 = blockIdx.y * 16;
  const int col = n0 + l16;

  v8f acc[MB];
  const v8f vzero = {};
#pragma unroll
  for (int i = 0; i < MB; ++i) acc[i] = vzero;

  // Per-lane base pointers; all subsequent accesses use constant offsets.
  const float* __restrict__ abase[MB];
#pragma unroll
  for (int i = 0; i < MB; ++i)
    abase[i] = A + (size_t)(m0 + i * 16 + l16) * K + (half << 1);
  const float* __restrict__ bbase = B + (size_t)(half << 1) * NCOLS + col;

  constexpr int STEPS = K / 4;
  v2f    bq[2];
  float2 aq[2][MB];

  // Prefetch step 0.
  bq[0].x = bbase[0];
  bq[0].y = bbase[NCOLS];
#pragma unroll
  for (int i = 0; i < MB; ++i)
    aq[0][i] = *reinterpret_cast<const float2*>(abase[i]);

#pragma unroll
  for (int kk = 0; kk < STEPS; ++kk) {
    const int cur = kk & 1;
    const int nxt = cur ^ 1;
    if (kk + 1 < STEPS) {
      const int ko = (kk + 1) * 4;                  // compile-time constant
      bq[nxt].x = bbase[(size_t)(ko + 0) * NCOLS];  // imm-offset load
      bq[nxt].y = bbase[(size_t)(ko + 1) * NCOLS];
#pragma unroll
      for (int i = 0; i < MB; ++i)
        aq[nxt][i] = *reinterpret_cast<const float2*>(abase[i] + ko);
    }
    const v2f b = bq[cur];
#pragma unroll
    for (int i = 0; i < MB; ++i) {
      v2f a;
      a.x = RELU ? fmaxf(aq[cur][i].x, 0.0f) : aq[cur][i].x;
      a.y = RELU ? fmaxf(aq[cur][i].y, 0.0f) : aq[cur][i].y;
      acc[i] = __builtin_amdgcn_wmma_f32_16x16x4_f32(
          false, a, false, b, (short)0, acc[i], false, false);
    }
  }

  const float bv = bias ? bias[col] : 0.0f;
  float* __restrict__ cbase = C + (size_t)(m0 + (half << 3)) * NCOLS + col;
#pragma unroll
  for (int i = 0; i < MB; ++i) {
#pragma unroll
    for (int r = 0; r < 8; ++r) {
      cbase[(size_t)(i * 16 + r) * NCOLS] = acc[i][r] + bv;  // imm-offset store
    }
  }
}

// ---------------------------------------------------------------------------
// Utility kernels
// ---------------------------------------------------------------------------
__global__ void fill_f32(float* __restrict__ p, float v, long long n) {
  long long i = (long long)blockIdx.x * blockDim.x + threadIdx.x;
  long long stride = (long long)gridDim.x * blockDim.x;
  for (; i < n; i += stride) p[i] = v;
}

__global__ void degree_edges(const long long* __restrict__ dst,
                             float* __restrict__ deg, int E) {
  int e = blockIdx.x * blockDim.x + threadIdx.x;
  if (e < E) atomicAdd(&deg[(int)dst[e]], 1.0f);
}

__global__ void rsqrt_inplace(float* __restrict__ d, int n) {
  int i = blockIdx.x * blockDim.x + threadIdx.x;
  if (i < n) d[i] = rsqrtf(d[i]);   // deg >= 1 always (self loops folded in)
}

// Zero-pad W (K x Nin) -> Wp (K x Npad) and b (Nin) -> bp (Npad)
__global__ void pad_wb(const float* __restrict__ W, const float* __restrict__ b,
                       float* __restrict__ Wp, float* __restrict__ bp,
                       int K, int Nin, int Npad) {
  int i = blockIdx.x * blockDim.x + threadIdx.x;
  if (i < Npad) bp[i] = (i < Nin) ? b[i] : 0.0f;
  if (i < K * Npad) {
    int r = i / Npad, c = i % Npad;
    Wp[i] = (c < Nin) ? W[r * Nin + c] : 0.0f;
  }
}

// ---------------------------------------------------------------------------
// Edge scatter: out[dst] += t[src] * dinv[src]*dinv[dst]
// One thread per (edge, 4-feature chunk); float4 gather + 4 f32 atomics.
// ---------------------------------------------------------------------------
__global__ void scatter_edges(const float* __restrict__ t,
                              const long long* __restrict__ src,
                              const long long* __restrict__ dst,
                              const float* __restrict__ dinv,
                              float* __restrict__ out, int E, int H) {
  const int chunks = H >> 2;
  long long gid = (long long)blockIdx.x * blockDim.x + threadIdx.x;
  if (gid >= (long long)E * chunks) return;
  const int e = (int)(gid / chunks);
  const int c = (int)(gid % chunks) << 2;
  const int s = (int)src[e];
  const int d = (int)dst[e];
  const float nrm = dinv[s] * dinv[d];
  const float4 v = *reinterpret_cast<const float4*>(t + (long long)s * H + c);
  float* o = out + (long long)d * H + c;
  atomicAdd(o + 0, v.x * nrm);
  atomicAdd(o + 1, v.y * nrm);
  atomicAdd(o + 2, v.z * nrm);
  atomicAdd(o + 3, v.w * nrm);
}

// ---------------------------------------------------------------------------
// Fused: agg += t*dinv^2 (self loop) + bias ; ReLU ; LayerNorm(g, be) in place.
// One wave32 per row, H in {64, 128}, wave shuffle reductions.
// ---------------------------------------------------------------------------
__global__ void selfloop_bias_relu_ln(float* __restrict__ agg,
                                      const float* __restrict__ t,
                                      const float* __restrict__ dinv,
                                      const float* __restrict__ bias,
                                      const float* __restrict__ g,
                                      const float* __restrict__ be,
                                      int N, int H) {
  const int node = blockIdx.x * (blockDim.x >> 5) + (threadIdx.x >> 5);
  const int lane = threadIdx.x & 31;
  if (node >= N) return;
  const float dn = dinv[node];
  const float self = dn * dn;
  const int per = H >> 5;           // 2 (H=64) or 4 (H=128)
  float vals[4];
  float mu = 0.0f;
  for (int i = 0; i < per; ++i) {
    const int f = lane + (i << 5);
    float v = agg[(long long)node * H + f]
            + t[(long long)node * H + f] * self + bias[f];
    v = fmaxf(v, 0.0f);
    vals[i] = v;
    mu += v;
  }
  for (int off = 16; off > 0; off >>= 1) mu += __shfl_xor(mu, off, 32);
  mu /= (float)H;
  float var = 0.0f;
  for (int i = 0; i < per; ++i) { const float d = vals[i] - mu; var += d * d; }
  for (int off = 16; off > 0; off >>= 1) var += __shfl_xor(var, off, 32);
  var /= (float)H;
  const float rs = rsqrtf(var + LN_EPS);
  for (int i = 0; i < per; ++i) {
    const int f = lane + (i << 5);
    agg[(long long)node * H + f] = (vals[i] - mu) * rs * g[f] + be[f];
  }
}

// emb = agg + t*dinv^2 + b3  (no ReLU/LN; elementwise, in place on agg)
__global__ void emb_finalize(float* __restrict__ out, const float* __restrict__ t,
                             const float* __restrict__ dinv,
                             const float* __restrict__ bias, int N, int H) {
  long long i = (long long)blockIdx.x * blockDim.x + threadIdx.x;
  if (i >= (long long)N * H) return;
  const int n = (int)(i / H);
  const int f = (int)(i % H);
  const float dn = dinv[n];
  out[i] += t[i] * dn * dn + bias[f];
}

// ---------------------------------------------------------------------------
// log_softmax over C classes (C <= 64); logits have row stride `ldl` (padded),
// output is compact N x C.  One wave32 per row.
// ---------------------------------------------------------------------------
__global__ void log_softmax_rows(const float* __restrict__ logits,
                                 float* __restrict__ out, int N, int C,
                                 int ldl) {
  const int node = blockIdx.x * (blockDim.x >> 5) + (threadIdx.x >> 5);
  const int lane = threadIdx.x & 31;
  if (node >= N) return;
  const float* row = logits + (long long)node * ldl;
  const bool ok0 = lane < C;
  const bool ok1 = (lane + 32) < C;
  const float v0 = ok0 ? row[lane] : -INFINITY;
  const float v1 = ok1 ? row[lane + 32] : -INFINITY;
  float mx = fmaxf(v0, v1);
  for (int off = 16; off > 0; off >>= 1) mx = fmaxf(mx, __shfl_xor(mx, off, 32));
  float s = (ok0 ? __expf(v0 - mx) : 0.0f) + (ok1 ? __expf(v1 - mx) : 0.0f);
  for (int off = 16; off > 0; off >>= 1) s += __shfl_xor(s, off, 32);
  const float lse = mx + __logf(s);
  float* orow = out + (long long)node * C;
  if (ok0) orow[lane] = v0 - lse;
  if (ok1) orow[lane + 32] = v1 - lse;
}

// ---------------------------------------------------------------------------
// Orchestration
// ---------------------------------------------------------------------------
extern "C" void kernel_launch(void* const* d_in, const int* in_sizes, int n_in,
                              void* d_out, int out_size, void* d_ws, size_t ws_size,
                              hipStream_t stream) {
  const int DIN = 128, H1 = 128, H2 = 64, OUT = 40, OUTP = 48;  // OUT padded
  const int N = in_sizes[0] / DIN;          // 50000
  const int E = in_sizes[1] / 2;            // 800000
  const int MB = 5;                         // 50000 = 625 * (5*16)

  const float*     x    = (const float*)d_in[0];
  const long long* eidx = (const long long*)d_in[1];   // int64 edge_index [2,E]
  const long long* src  = eidx;
  const long long* dst  = eidx + E;
  const float* W1  = (const float*)d_in[2];
  const float* b1  = (const float*)d_in[3];
  const float* W2  = (const float*)d_in[4];
  const float* b2  = (const float*)d_in[5];
  const float* W3  = (const float*)d_in[6];
  const float* b3  = (const float*)d_in[7];
  const float* g1  = (const float*)d_in[8];
  const float* be1 = (const float*)d_in[9];
  const float* g2  = (const float*)d_in[10];
  const float* be2 = (const float*)d_in[11];
  const float* Wp1 = (const float*)d_in[12];
  const float* bp1 = (const float*)d_in[13];
  const float* Wp2 = (const float*)d_in[14];
  const float* bp2 = (const float*)d_in[15];

  float* out   = (float*)d_out;             // [0,N*H2) = emb, then N*OUT logsm
  float* emb   = out;
  float* logsm = out + (long long)N * H2;

  // workspace: dinv (N) | bufA (N*128) | bufB (N*128) | Wp2pad (64*48) | bp2pad (48)
  float* dinv   = (float*)d_ws;
  float* bufA   = dinv + N;
  float* bufB   = bufA + (long long)N * 128;
  float* Wp2pad = bufB + (long long)N * 128;
  float* bp2pad = Wp2pad + (long long)H2 * OUTP;

  const int TB = 256;
  const int waveRows = TB / 32;
  dim3 blk32(32), blkTB(TB);
  auto cdiv = [](long long a, long long b) { return (int)((a + b - 1) / b); };

  // --- degree / dinv (self-loop folded in as init=1) ---
  fill_f32<<<cdiv(N, TB), blkTB, 0, stream>>>(dinv, 1.0f, N);
  degree_edges<<<cdiv(E, TB), blkTB, 0, stream>>>(dst, dinv, E);
  rsqrt_inplace<<<cdiv(N, TB), blkTB, 0, stream>>>(dinv, N);

  // --- pad Wp2 / bp2 to 48 cols for guard-free WMMA ---
  pad_wb<<<cdiv(H2 * OUTP, TB), blkTB, 0, stream>>>(Wp2, bp2, Wp2pad, bp2pad,
                                                    H2, OUT, OUTP);

  // --- Layer 1: t1 = x@W1 -> bufA ; scatter -> bufB ; fused post -> h1(bufB)
  wmma_gemm_f32<MB, 128, 128, false>
      <<<dim3(N / (MB * 16), H1 / 16), blk32, 0, stream>>>(x, W1, nullptr, bufA);
  fill_f32<<<cdiv((long long)N * H1, TB), blkTB, 0, stream>>>(bufB, 0.0f,
                                                             (long long)N * H1);
  scatter_edges<<<cdiv((long long)E * (H1 / 4), TB), blkTB, 0, stream>>>(
      bufA, src, dst, dinv, bufB, E, H1);
  selfloop_bias_relu_ln<<<cdiv(N, waveRows), blkTB, 0, stream>>>(
      bufB, bufA, dinv, b1, g1, be1, N, H1);

  // --- Layer 2: t2 = h1@W2 -> bufA[0,N*64) ; agg2 at bufA+N*64 ; h2 there
  float* t2   = bufA;
  float* agg2 = bufA + (long long)N * H2;
  wmma_gemm_f32<MB, 128, 64, false>
      <<<dim3(N / (MB * 16), H2 / 16), blk32, 0, stream>>>(bufB, W2, nullptr, t2);
  fill_f32<<<cdiv((long long)N * H2, TB), blkTB, 0, stream>>>(agg2, 0.0f,
                                                             (long long)N * H2);
  scatter_edges<<<cdiv((long long)E * (H2 / 4), TB), blkTB, 0, stream>>>(
      t2, src, dst, dinv, agg2, E, H2);
  selfloop_bias_relu_ln<<<cdiv(N, waveRows), blkTB, 0, stream>>>(
      agg2, t2, dinv, b2, g2, be2, N, H2);

  // --- Layer 3: t3 = h2@W3 -> bufB ; scatter directly into emb (d_out)
  wmma_gemm_f32<MB, 64, 64, false>
      <<<dim3(N / (MB * 16), H2 / 16), blk32, 0, stream>>>(agg2, W3, nullptr, bufB);
  fill_f32<<<cdiv((long long)N * H2, TB), blkTB, 0, stream>>>(emb, 0.0f,
                                                             (long long)N * H2);
  scatter_edges<<<cdiv((long long)E * (H2 / 4), TB), blkTB, 0, stream>>>(
      bufB, src, dst, dinv, emb, E, H2);
  emb_finalize<<<cdiv((long long)N * H2, TB), blkTB, 0, stream>>>(
      emb, bufB, dinv, b3, N, H2);

  // --- Head: p1 = relu(emb)@Wp1 + bp1 -> bufA ; logits = p1@Wp2pad + bp2pad
  wmma_gemm_f32<MB, 64, 64, true>
      <<<dim3(N / (MB * 16), H2 / 16), blk32, 0, stream>>>(emb, Wp1, bp1, bufA);
  wmma_gemm_f32<MB, 64, 48, false>
      <<<dim3(N / (MB * 16), OUTP / 16), blk32, 0, stream>>>(bufA, Wp2pad,
                                                             bp2pad, bufB);

  // --- log_softmax (reads padded stride 48, first 40 cols) -> output region
  log_softmax_rows<<<cdiv(N, waveRows), blkTB, 0, stream>>>(bufB, logsm, N, OUT,
                                                            OUTP);
}